// MoESparseLayer_63926293233905
// MI455X (gfx1250) — compile-verified
//
#include <hip/hip_runtime.h>
#include <hip/hip_bf16.h>
#include <math.h>
#include <stdint.h>

// ---------------- problem constants ----------------
static constexpr int DD   = 1024;   // hidden dim
static constexpr int EE   = 8;      // experts
static constexpr int HH   = 4096;   // expert hidden
static constexpr int NTOK = 4096;   // tokens (4*1024)

// ---------------- GEMM tiling ----------------
static constexpr int BM   = 64;     // block M tile
static constexpr int BN   = 128;    // block N tile
static constexpr int KC   = 32;     // K chunk staged in LDS
static constexpr int LDA  = 36;     // A row stride (floats): 144B rows, 16B aligned, conflict-free b64 frags
static constexpr int LDBT = 34;     // transposed-B row stride (floats): 17n mod 32 distinct -> conflict-free

typedef float v2f __attribute__((ext_vector_type(2)));
typedef float v8f __attribute__((ext_vector_type(8)));

// ---------------- CDNA5 async global->LDS copy ----------------
// builtin signature (probe-confirmed):
//   void __builtin_amdgcn_global_load_async_to_lds_b128(int4 AS1*, int4 AS3*, imm offset, imm cpol)
typedef __attribute__((__vector_size__(4 * sizeof(int)))) int vi4;
typedef __attribute__((address_space(1))) vi4  g_vi4;
typedef __attribute__((address_space(3))) vi4  l_vi4;
typedef __attribute__((address_space(3))) void las_void;
typedef __attribute__((address_space(3))) char las_char;

__device__ __forceinline__ void async_ld_b128(const void* g, void* l) {
#if __has_builtin(__builtin_amdgcn_global_load_async_to_lds_b128)
    __builtin_amdgcn_global_load_async_to_lds_b128(
        (g_vi4*)(uintptr_t)g, (l_vi4*)(las_void*)l, 0, 0);
#else
    const uint32_t loff = (uint32_t)(uintptr_t)(las_char*)l;
    asm volatile("global_load_async_to_lds_b128 %0, %1, off"
                 :: "v"(loff), "v"(g) : "memory");
#endif
}

template <int N>
__device__ __forceinline__ void wait_asynccnt() {
#if __has_builtin(__builtin_amdgcn_s_wait_asynccnt)
    __builtin_amdgcn_s_wait_asynccnt((unsigned short)N);
#else
    asm volatile("s_wait_asynccnt %0" :: "n"(N) : "memory");
#endif
}

// =====================================================================
// Gating: one wave (32 lanes) per token. logits = x . Wg + bg, softmax,
// top-2, push (token, gateWeight) into per-expert bucket via atomics.
// =====================================================================
__global__ __launch_bounds__(256)
void moe_gate(const float* __restrict__ x, const float* __restrict__ Wg,
              const float* __restrict__ bg, int* __restrict__ counts,
              int* __restrict__ toks, float* __restrict__ wgts) {
    const int lane = threadIdx.x & 31;
    const int wave = threadIdx.x >> 5;
    const int t = blockIdx.x * 8 + wave;
    if (t >= NTOK) return;

    float acc[EE];
#pragma unroll
    for (int e = 0; e < EE; ++e) acc[e] = 0.0f;

    const float* xr = x + (size_t)t * DD;
    for (int d = lane; d < DD; d += 32) {
        const float xv = xr[d];
        const float* wr = Wg + (size_t)d * EE;
#pragma unroll
        for (int e = 0; e < EE; ++e) acc[e] += xv * wr[e];
    }
#pragma unroll
    for (int off = 16; off > 0; off >>= 1) {
#pragma unroll
        for (int e = 0; e < EE; ++e) acc[e] += __shfl_xor(acc[e], off, 32);
    }

    if (lane == 0) {
        float lg[EE];
        float mx = -3.0e38f;
#pragma unroll
        for (int e = 0; e < EE; ++e) { lg[e] = acc[e] + bg[e]; mx = fmaxf(mx, lg[e]); }
        float s = 0.0f;
#pragma unroll
        for (int e = 0; e < EE; ++e) { lg[e] = expf(lg[e] - mx); s += lg[e]; }
        const float inv = 1.0f / s;

        int i1 = 0; float p1 = lg[0];
#pragma unroll
        for (int e = 1; e < EE; ++e) if (lg[e] > p1) { p1 = lg[e]; i1 = e; }
        int i2 = (i1 == 0) ? 1 : 0; float p2 = lg[i2];
#pragma unroll
        for (int e = 0; e < EE; ++e)
            if (e != i1 && lg[e] > p2) { p2 = lg[e]; i2 = e; }

        int pos1 = atomicAdd(&counts[i1], 1);
        toks[i1 * NTOK + pos1] = t;
        wgts[i1 * NTOK + pos1] = p1 * inv;
        int pos2 = atomicAdd(&counts[i2], 1);
        toks[i2 * NTOK + pos2] = t;
        wgts[i2 * NTOK + pos2] = p2 * inv;
    }
}

// exclusive prefix over the 8 expert counts (compacts the h-buffer rows)
__global__ void moe_prefix(const int* __restrict__ counts, int* __restrict__ offsets) {
    if (threadIdx.x == 0 && blockIdx.x == 0) {
        int s = 0;
#pragma unroll
        for (int e = 0; e < EE; ++e) { offsets[e] = s; s += counts[e]; }
    }
}

// =====================================================================
// Grouped expert GEMM on V_WMMA_F32_16X16X4_F32.
//  - A (activations, gathered): double-buffered GLOBAL_LOAD_ASYNC_TO_LDS_B128
//    DMA (ASYNCcnt), natural [m][k] layout -> b64 fragments.
//  - B (weights): software-pipelined register staging with VALU transpose
//    into a single sBt[n][k] buffer -> b64 fragments, zero repack movs.
//   L2 == false : h = gelu(X_e[gathered] @ W1[e] + b1[e])   K=D,  N=H
//   L2 == true  : out += gate_w * (h_e @ W2[e] + b2[e])     K=H,  N=D
// Block: 256 threads / 8 waves, tile BM x BN. Wave w owns a 16x64 strip
// (m0=(w>>1)*16, n0=(w&1)*64) -> 4 v8f accumulators.
// =====================================================================
template <bool L2>
__global__ __launch_bounds__(256)
void moe_ffn(const float* __restrict__ Ain,      // L2 ? hbuf : x
             const float* __restrict__ W,        // [E][KD][ND] row-major
             const float* __restrict__ bias,     // [E][ND]
             const int*   __restrict__ toks,     // [E][NTOK]
             const float* __restrict__ wgts,     // [E][NTOK]
             const int*   __restrict__ counts,   // [E]
             const int*   __restrict__ offsets,  // [E]
             float*       __restrict__ outp)     // L2 ? out[N][D] : hbuf[rows][H]
{
    constexpr int KD = L2 ? HH : DD;
    constexpr int ND = L2 ? DD : HH;

    const int e     = blockIdx.z;
    const int Me    = counts[e];
    const int mbase = blockIdx.y * BM;
    if (mbase >= Me) return;
    const int nbase = blockIdx.x * BN;
    const int rbase = offsets[e];

    __shared__ float sA[2][BM][LDA];     // 2*64*36*4  = 18.4 KB (async DMA target)
    __shared__ float sBt[BN][LDBT];      // 128*34*4   = 17.4 KB (transposed weights)

    const int tid  = threadIdx.x;
    const int lane = tid & 31;
    const int wave = tid >> 5;
    const int m0 = (wave >> 1) * 16;     // 0,16,32,48
    const int n0 = (wave & 1) * 64;      // 0,64

    v8f acc[4] = {};

    const float* Wbase = W + (size_t)e * KD * ND;

    const int mrow  = m0 + (lane & 15);  // A fragment row
    const int khalf = (lane >> 4) << 1;  // 0 or 2
    const int ncol  = lane & 15;

    // A async staging: 2 x b128 per wave per chunk (4 rows x 8 lanes of 16B each)
    const int arow0 = wave * 8 + (lane >> 3);
    const int aseg  = (lane & 7) << 2;

    auto stageA = [&](int p, int k0) {
#pragma unroll
        for (int it = 0; it < 2; ++it) {
            const int row = arow0 + it * 4;
            const int mg  = mbase + row;
            const int rc  = (mg < Me) ? mg : (Me - 1);   // clamp; results discarded
            const float* g;
            if (L2) g = Ain + (size_t)(rbase + rc) * KD + (k0 + aseg);
            else    g = Ain + (size_t)toks[e * NTOK + rc] * KD + (k0 + aseg);
            async_ld_b128(g, &sA[p][row][aseg]);
        }
    };

    // B register staging: 4 x float4 per thread per chunk (KC x BN = 4096 floats)
    float4 breg[4];
    auto loadB = [&](int k0) {
#pragma unroll
        for (int it = 0; it < 4; ++it) {
            const int idx = tid + it * 256;       // 0..1023
            const int k   = idx >> 5;             // 0..31
            const int n4  = (idx & 31) << 2;      // 0..124
            breg[it] = *reinterpret_cast<const float4*>(
                Wbase + (size_t)(k0 + k) * ND + (nbase + n4));
        }
    };
    auto storeB = [&]() {
#pragma unroll
        for (int it = 0; it < 4; ++it) {
            const int idx = tid + it * 256;
            const int k   = idx >> 5;
            const int n4  = (idx & 31) << 2;
            sBt[n4 + 0][k] = breg[it].x;
            sBt[n4 + 1][k] = breg[it].y;
            sBt[n4 + 2][k] = breg[it].z;
            sBt[n4 + 3][k] = breg[it].w;
        }
    };

    loadB(0);
    stageA(0, 0);
    int buf = 0;
    for (int k0 = 0; k0 < KD; k0 += KC) {
        const bool hasNext = (k0 + KC) < KD;
        storeB();                                   // current chunk's weights -> LDS (transposed)
        if (hasNext) {
            loadB(k0 + KC);                         // next chunk's weights -> regs (latency hidden)
            stageA(buf ^ 1, k0 + KC);               // next chunk's activations -> DMA
            wait_asynccnt<2>();                     // current A buffer complete (in-order)
        } else {
            wait_asynccnt<0>();
        }
        __syncthreads();

#pragma unroll
        for (int kk = 0; kk < KC; kk += 4) {
            const v2f a = *reinterpret_cast<const v2f*>(&sA[buf][mrow][kk + khalf]);
#pragma unroll
            for (int j = 0; j < 4; ++j) {
                const v2f b = *reinterpret_cast<const v2f*>(
                    &sBt[n0 + j * 16 + ncol][kk + khalf]);
                acc[j] = __builtin_amdgcn_wmma_f32_16x16x4_f32(
                    false, a, false, b, (short)0, acc[j], false, false);
            }
        }
        __syncthreads();                            // all reads done; buffers reusable
        buf ^= 1;
    }

    // ---- epilogue. C layout: vgpr r, lanes0-15 -> M=r, lanes16-31 -> M=r+8, N=lane&15
    const int mhalf = 8 * (lane >> 4);
#pragma unroll
    for (int j = 0; j < 4; ++j) {
        const int n  = nbase + n0 + j * 16 + ncol;
        const float bv = bias[e * ND + n];
#pragma unroll
        for (int r = 0; r < 8; ++r) {
            const int m = mbase + m0 + mhalf + r;
            if (m < Me) {
                const float v = acc[j][r] + bv;
                if (L2) {
                    const float gw = wgts[e * NTOK + m];
                    const int  tok = toks[e * NTOK + m];
                    atomicAdd(&outp[(size_t)tok * DD + n], v * gw);
                } else {
                    const float g = 0.5f * v * (1.0f + erff(v * 0.70710678118654752f));
                    outp[(size_t)(rbase + m) * HH + n] = g;
                }
            }
        }
    }
}

// =====================================================================
// launch
// =====================================================================
extern "C" void kernel_launch(void* const* d_in, const int* in_sizes, int n_in,
                              void* d_out, int out_size, void* d_ws, size_t ws_size,
                              hipStream_t stream) {
    const float* x  = (const float*)d_in[0];
    const float* W1 = (const float*)d_in[1];
    const float* b1 = (const float*)d_in[2];
    const float* W2 = (const float*)d_in[3];
    const float* b2 = (const float*)d_in[4];
    const float* Wg = (const float*)d_in[5];
    const float* bg = (const float*)d_in[6];
    float* out = (float*)d_out;

    // workspace layout
    char* ws = (char*)d_ws;
    int*   counts  = (int*)ws;                                   // 8 ints
    int*   offsets = (int*)(ws + 32);                            // 8 ints
    int*   toks    = (int*)(ws + 64);                            // E*NTOK ints
    float* wgts    = (float*)(ws + 64 + (size_t)EE * NTOK * 4);  // E*NTOK floats
    size_t hoff    = (64 + 2 * (size_t)EE * NTOK * 4 + 255) & ~(size_t)255;
    float* hbuf    = (float*)(ws + hoff);                        // (N*TOPK) x H floats

    (void)hipMemsetAsync(counts, 0, 64, stream);
    (void)hipMemsetAsync(d_out, 0, (size_t)out_size * sizeof(float), stream);

    // 1) gating + routing
    moe_gate<<<NTOK / 8, 256, 0, stream>>>(x, Wg, bg, counts, toks, wgts);
    // 2) bucket offsets
    moe_prefix<<<1, 32, 0, stream>>>(counts, offsets);
    // 3) layer-1 grouped GEMM + GELU  (worst-case grid; empty m-tiles exit early)
    dim3 g1(HH / BN, NTOK / BM, EE);
    moe_ffn<false><<<g1, 256, 0, stream>>>(x, W1, b1, toks, wgts, counts, offsets, hbuf);
    // 4) layer-2 grouped GEMM + gated scatter-add
    dim3 g2(DD / BN, NTOK / BM, EE);
    moe_ffn<true><<<g2, 256, 0, stream>>>(hbuf, W2, b2, toks, wgts, counts, offsets, out);
}